// GSM_ones_only_55387898249345
// MI455X (gfx1250) — compile-verified
//
#include <hip/hip_runtime.h>

typedef __attribute__((ext_vector_type(2))) float v2f;
typedef __attribute__((ext_vector_type(8))) float v8f;

#define B_DIM 256
#define N_DIM 512
#define M_DIM 1024

// One wave32 per (batch b, 16-row n-tile). K-loop over M in steps of 4 using
// V_WMMA_F32_16X16X4_F32:  D = A(16x4 softmax weights) x B(4x16: [x | 1 | 0...])
// => D[i,0] = numerator, D[i,1] = denominator, accumulated in f32 across all M.
__global__ __launch_bounds__(256) void gsm_wmma_f32_kernel(
    const float* __restrict__ x,       // [B, M]
    const float* __restrict__ logits,  // [N, M]
    const float* __restrict__ gumbel,  // [B, N, M]
    float* __restrict__ out)           // [B, N]
{
  const int lane = threadIdx.x & 31;
  const int wave = threadIdx.x >> 5;
  const int tile = blockIdx.x * 8 + wave;       // 8192 tiles total
  const int b    = tile >> 5;                   // tile / (N/16)
  const int n0   = (tile & 31) << 4;            // n-tile base (16 rows)

  const int row = n0 + (lane & 15);             // this lane's A-matrix row
  const int kp  = (lane >> 4) << 1;             // k-pair: 0 (lanes 0-15) or 2 (lanes 16-31)
  const int col = lane & 15;                    // this lane's B-matrix column

  const float* gp = gumbel + ((size_t)b * N_DIM + row) * M_DIM + kp;
  const float* lp = logits + (size_t)row * M_DIM + kp;
  const float* xp = x + (size_t)b * M_DIM + kp;

  v8f c = {};  // D/C accumulator: col0 = num, col1 = den

  #pragma unroll 8
  for (int m0 = 0; m0 < M_DIM; m0 += 4) {
    v2f g  = *(const v2f*)(gp + m0);   // gumbel[b, row, m0+kp .. +1]
    v2f l  = *(const v2f*)(lp + m0);   // logits[row, m0+kp .. +1]  (L2 resident)
    v2f xv = *(const v2f*)(xp + m0);   // x[b, m0+kp .. +1]         (L0 resident)

    // A-matrix fragment: softmax weights (f32 exp, no overflow: z <= ~19)
    v2f a;
    a.x = __expf(g.x + l.x);
    a.y = __expf(g.y + l.y);

    // B-matrix fragment: column 0 carries x, column 1 carries ones
    v2f bb;
    bb.x = (col == 0) ? xv.x : ((col == 1) ? 1.0f : 0.0f);
    bb.y = (col == 0) ? xv.y : ((col == 1) ? 1.0f : 0.0f);

    // 8 args: (neg_a, A, neg_b, B, c_mod, C, reuse_a, reuse_b)
    c = __builtin_amdgcn_wmma_f32_16x16x4_f32(
        /*neg_a=*/false, a, /*neg_b=*/false, bb,
        /*c_mod=*/(short)0, c, /*reuse_a=*/false, /*reuse_b=*/false);
  }

  // D layout (32-bit C/D 16x16): VGPR r, lanes 0-15 -> M=r, N=lane;
  // lanes 16-31 -> M=8+r, N=lane-16. Numerator lives in col 0 (lanes 0,16),
  // denominator in col 1 (lanes 1,17): pull it over with ds_bpermute(lane+1).
  #pragma unroll
  for (int r = 0; r < 8; ++r) {
    float num = c[r];
    int   di  = __builtin_amdgcn_ds_bpermute(((lane + 1) & 31) << 2,
                                             __float_as_int(num));
    float den = __int_as_float(di);
    if (col == 0) {
      int orow = n0 + r + ((lane >> 4) << 3);
      out[(size_t)b * N_DIM + orow] = num / den;
    }
  }
}

extern "C" void kernel_launch(void* const* d_in, const int* in_sizes, int n_in,
                              void* d_out, int out_size, void* d_ws, size_t ws_size,
                              hipStream_t stream) {
  const float* x      = (const float*)d_in[0];  // input  [B, M]
  const float* logits = (const float*)d_in[1];  // logits [N, M]
  const float* gumbel = (const float*)d_in[2];  // gumbel [B, N, M]
  float* out = (float*)d_out;                   // [B, N]

  // 8192 wave-tiles, 8 waves (256 threads) per block -> 1024 blocks
  dim3 grid(B_DIM * (N_DIM / 16) / 8);
  dim3 block(256);
  gsm_wmma_f32_kernel<<<grid, block, 0, stream>>>(x, logits, gumbel, out);
}